// SymbolCommModule_48301202211076
// MI455X (gfx1250) — compile-verified
//
#include <hip/hip_runtime.h>
#include <hip/hip_bf16.h>
#include <math.h>

// ---------------------------------------------------------------------------
// SymbolComm module for MI455X (gfx1250, wave32).
//  k1: chunked pooling partials   (read 256MB, deterministic tree reduce)
//  k2: finalize pooled mean [8,2048]
//  k3: fused MLP brain, all 4 GEMMs via V_WMMA_F32_16X16X4_F32 (one WG, 16 waves)
//  k4: float4 residual broadcast  (read 256MB + write 256MB)
// ---------------------------------------------------------------------------

typedef float v2f __attribute__((ext_vector_type(2)));
typedef float v8f __attribute__((ext_vector_type(8)));

#define WMMA_F32X4(a, b, c) \
    __builtin_amdgcn_wmma_f32_16x16x4_f32(false, (a), false, (b), (short)0, (c), false, false)

static constexpr int B = 8, S = 4096, H = 2048;
static constexpr int V = 32, L = 8, E = 64;
static constexpr int POOL_CHUNKS = 64;                // S split into 64 chunks of 64 rows
static constexpr float LN_EPS = 1e-5f;

// ---------------- k1: pooling partial sums ---------------------------------
__global__ __launch_bounds__(256)
void pool_partial_kernel(const float* __restrict__ hidden, float* __restrict__ partials) {
    const int b    = blockIdx.x;   // 8
    const int hblk = blockIdx.y;   // 8
    const int ch   = blockIdx.z;   // 64
    const int h = hblk * 256 + threadIdx.x;
    const float* p = hidden + ((size_t)b * S + (size_t)ch * 64) * H + h;
    float s = 0.0f;
#pragma unroll 8
    for (int i = 0; i < 64; ++i) s += p[(size_t)i * H];
    partials[((size_t)ch * B + b) * H + h] = s;
}

// ---------------- k2: finalize pooled mean ---------------------------------
__global__ __launch_bounds__(256)
void pool_finalize_kernel(const float* __restrict__ partials, float* __restrict__ pooled) {
    const int idx = blockIdx.x * 256 + threadIdx.x;   // 0..16383
    float s = 0.0f;
#pragma unroll 8
    for (int ch = 0; ch < POOL_CHUNKS; ++ch) s += partials[(size_t)ch * (B * H) + idx];
    pooled[idx] = s * (1.0f / (float)S);
}

// ---------------- k3: fused encoder/gumbel/decoder brain --------------------
__device__ __forceinline__ float gelu_exact(float x) {
    return 0.5f * x * (1.0f + erff(x * 0.70710678118654752440f));
}

__global__ __launch_bounds__(512)
void brain_kernel(const float* __restrict__ pooled,     // [8,2048]
                  const float* __restrict__ gumbel_u,   // [8,8,32]
                  const float* __restrict__ enc_w1, const float* __restrict__ enc_b1,
                  const float* __restrict__ ln1_g,  const float* __restrict__ ln1_b,
                  const float* __restrict__ enc_w2, const float* __restrict__ enc_b2,
                  const float* __restrict__ embed,
                  const float* __restrict__ dec_w1, const float* __restrict__ dec_b1,
                  const float* __restrict__ ln2_g,  const float* __restrict__ ln2_b,
                  const float* __restrict__ dec_w2, const float* __restrict__ dec_b2,
                  float* __restrict__ decoded)          // [8,2048]
{
    __shared__ float sH[8][256];   // encoder hidden (post LN+GELU)
    __shared__ float sL[8][256];   // logits -> straight-through symbols
    __shared__ float sE[8][512];   // flattened embeddings
    __shared__ float sD[8][256];   // decoder hidden (post LN+GELU)
    __shared__ float sMu[8], sRs[8];

    const int tid  = threadIdx.x;
    const int wave = tid >> 5;        // 16 waves
    const int lane = tid & 31;
    const int m    = lane & 15;       // A row / B column within the 16-tile
    const int mr   = m & 7;           // clamped row (rows 8..15 are zero padding)
    const int kh   = (lane >> 4) << 1;  // K sub-pair: lanes 0-15 -> K0/K1, 16-31 -> K2/K3
    const float mmask = (m < 8) ? 1.0f : 0.0f;

    // ===== GEMM1: sH = pooled[8,2048] @ enc_w1.T   (K=2048, N=256) =====
    {
        const int n0 = wave * 16;
        const float* Bp = enc_w1 + (size_t)(n0 + m) * 2048;
        const float* Ap = pooled + (size_t)mr * 2048;
        v8f acc = {0.f, 0.f, 0.f, 0.f, 0.f, 0.f, 0.f, 0.f};
#pragma unroll 4
        for (int k0 = 0; k0 < 2048; k0 += 4) {
            v2f a, b;
            a.x = Ap[k0 + kh] * mmask;
            a.y = Ap[k0 + kh + 1] * mmask;
            b.x = Bp[k0 + kh];
            b.y = Bp[k0 + kh + 1];
            acc = WMMA_F32X4(a, b, acc);
        }
        if (lane < 16) {
#pragma unroll
            for (int r = 0; r < 8; ++r) sH[r][n0 + lane] = acc[r];
        }
    }
    __syncthreads();

    // bias + LayerNorm1 + GELU
    if (tid < 256) {
        float bb = enc_b1[tid];
        for (int r = 0; r < 8; ++r) sH[r][tid] += bb;
    }
    __syncthreads();
    if (tid < 8) {
        float mu = 0.f;
        for (int j = 0; j < 256; ++j) mu += sH[tid][j];
        mu *= (1.0f / 256.0f);
        float var = 0.f;
        for (int j = 0; j < 256; ++j) { float d = sH[tid][j] - mu; var += d * d; }
        var *= (1.0f / 256.0f);
        sMu[tid] = mu; sRs[tid] = rsqrtf(var + LN_EPS);
    }
    __syncthreads();
    if (tid < 256) {
        float g = ln1_g[tid], bb = ln1_b[tid];
        for (int r = 0; r < 8; ++r) {
            float x = (sH[r][tid] - sMu[r]) * sRs[r] * g + bb;
            sH[r][tid] = gelu_exact(x);
        }
    }
    __syncthreads();

    // ===== GEMM2: sL = sH[8,256] @ enc_w2.T + enc_b2   (K=256, N=256) =====
    {
        const int n0 = wave * 16;
        const float* Bp = enc_w2 + (size_t)(n0 + m) * 256;
        v8f acc = {0.f, 0.f, 0.f, 0.f, 0.f, 0.f, 0.f, 0.f};
#pragma unroll 4
        for (int k0 = 0; k0 < 256; k0 += 4) {
            v2f a, b;
            a.x = sH[mr][k0 + kh] * mmask;
            a.y = sH[mr][k0 + kh + 1] * mmask;
            b.x = Bp[k0 + kh];
            b.y = Bp[k0 + kh + 1];
            acc = WMMA_F32X4(a, b, acc);
        }
        if (lane < 16) {
            float bias = enc_b2[n0 + lane];
#pragma unroll
            for (int r = 0; r < 8; ++r) sL[r][n0 + lane] = acc[r] + bias;
        }
    }
    __syncthreads();

    // ===== Gumbel-softmax + straight-through hard symbols (64 (b,l) pairs) =====
    if (tid < 64) {
        const int b = tid >> 3, l = tid & 7;
        float z[32];
        float zmax = -INFINITY;
        for (int v = 0; v < V; ++v) {
            float u = gumbel_u[((size_t)b * L + l) * V + v];
            float g = -logf(-logf(u));
            float zz = sL[b][l * V + v] + g;       // TAU = 1
            z[v] = zz;
            zmax = fmaxf(zmax, zz);
        }
        float s = 0.f;
        for (int v = 0; v < V; ++v) { z[v] = expf(z[v] - zmax); s += z[v]; }
        float inv = 1.0f / s;
        int idx = 0; float best = z[0];
        for (int v = 1; v < V; ++v) if (z[v] > best) { best = z[v]; idx = v; }
        for (int v = 0; v < V; ++v) {
            float ys = z[v] * inv;
            float hard = (v == idx) ? 1.0f : 0.0f;
            sL[b][l * V + v] = (hard - ys) + ys;   // match reference rounding order
        }
    }
    __syncthreads();

    // ===== embed lookup: sE[b][l*64+e] = sum_v sym[b,l,v] * embed[v,e] =====
    for (int o = tid; o < 8 * 512; o += 512) {
        const int b = o >> 9, col = o & 511;
        const int l = col >> 6, e = col & 63;
        float acc = 0.f;
#pragma unroll 8
        for (int v = 0; v < V; ++v) acc += sL[b][l * V + v] * embed[v * E + e];
        sE[b][col] = acc;
    }
    __syncthreads();

    // ===== GEMM3: sD = sE[8,512] @ dec_w1.T + dec_b1   (K=512, N=256) =====
    {
        const int n0 = wave * 16;
        const float* Bp = dec_w1 + (size_t)(n0 + m) * 512;
        v8f acc = {0.f, 0.f, 0.f, 0.f, 0.f, 0.f, 0.f, 0.f};
#pragma unroll 4
        for (int k0 = 0; k0 < 512; k0 += 4) {
            v2f a, b;
            a.x = sE[mr][k0 + kh] * mmask;
            a.y = sE[mr][k0 + kh + 1] * mmask;
            b.x = Bp[k0 + kh];
            b.y = Bp[k0 + kh + 1];
            acc = WMMA_F32X4(a, b, acc);
        }
        if (lane < 16) {
            float bias = dec_b1[n0 + lane];
#pragma unroll
            for (int r = 0; r < 8; ++r) sD[r][n0 + lane] = acc[r] + bias;
        }
    }
    __syncthreads();

    // LayerNorm2 + GELU
    if (tid < 8) {
        float mu = 0.f;
        for (int j = 0; j < 256; ++j) mu += sD[tid][j];
        mu *= (1.0f / 256.0f);
        float var = 0.f;
        for (int j = 0; j < 256; ++j) { float d = sD[tid][j] - mu; var += d * d; }
        var *= (1.0f / 256.0f);
        sMu[tid] = mu; sRs[tid] = rsqrtf(var + LN_EPS);
    }
    __syncthreads();
    if (tid < 256) {
        float g = ln2_g[tid], bb = ln2_b[tid];
        for (int r = 0; r < 8; ++r) {
            float x = (sD[r][tid] - sMu[r]) * sRs[r] * g + bb;
            sD[r][tid] = gelu_exact(x);
        }
    }
    __syncthreads();

    // ===== GEMM4: decoded[8,2048] = sD @ dec_w2.T + dec_b2 (K=256, N=2048) =====
    for (int t8 = 0; t8 < 8; ++t8) {
        const int n0 = (wave + t8 * 16) * 16;
        const float* Bp = dec_w2 + (size_t)(n0 + m) * 256;
        v8f acc = {0.f, 0.f, 0.f, 0.f, 0.f, 0.f, 0.f, 0.f};
#pragma unroll 4
        for (int k0 = 0; k0 < 256; k0 += 4) {
            v2f a, b;
            a.x = sD[mr][k0 + kh] * mmask;
            a.y = sD[mr][k0 + kh + 1] * mmask;
            b.x = Bp[k0 + kh];
            b.y = Bp[k0 + kh + 1];
            acc = WMMA_F32X4(a, b, acc);
        }
        if (lane < 16) {
            float bias = dec_b2[n0 + lane];
#pragma unroll
            for (int r = 0; r < 8; ++r) decoded[(size_t)r * H + n0 + lane] = acc[r] + bias;
        }
    }
}

// ---------------- k4: residual broadcast (float4 streaming) -----------------
__global__ __launch_bounds__(256)
void residual_kernel(const float4* __restrict__ hidden, const float4* __restrict__ decoded4,
                     float4* __restrict__ out, unsigned total4) {
    const unsigned stride = gridDim.x * blockDim.x;
    for (unsigned i = blockIdx.x * blockDim.x + threadIdx.x; i < total4; i += stride) {
        const unsigned hq = i & 511u;           // H/4 = 512
        const unsigned b  = i >> 21;            // S*H/4 = 2^21 per batch
        float4 hv = hidden[i];
        float4 dv = decoded4[b * 512u + hq];
        float4 o;
        o.x = hv.x + 0.1f * dv.x;
        o.y = hv.y + 0.1f * dv.y;
        o.z = hv.z + 0.1f * dv.z;
        o.w = hv.w + 0.1f * dv.w;
        out[i] = o;
    }
}

// ---------------------------------------------------------------------------
extern "C" void kernel_launch(void* const* d_in, const int* in_sizes, int n_in,
                              void* d_out, int out_size, void* d_ws, size_t ws_size,
                              hipStream_t stream) {
    const float* hidden   = (const float*)d_in[0];
    const float* gumbel_u = (const float*)d_in[1];
    const float* enc_w1   = (const float*)d_in[2];
    const float* enc_b1   = (const float*)d_in[3];
    const float* ln1_g    = (const float*)d_in[4];
    const float* ln1_b    = (const float*)d_in[5];
    const float* enc_w2   = (const float*)d_in[6];
    const float* enc_b2   = (const float*)d_in[7];
    const float* embed    = (const float*)d_in[8];
    const float* dec_w1   = (const float*)d_in[9];
    const float* dec_b1   = (const float*)d_in[10];
    const float* ln2_g    = (const float*)d_in[11];
    const float* ln2_b    = (const float*)d_in[12];
    const float* dec_w2   = (const float*)d_in[13];
    const float* dec_b2   = (const float*)d_in[14];
    (void)in_sizes; (void)n_in; (void)out_size; (void)ws_size;

    float* pooled   = (float*)d_ws;                  // 16384 floats
    float* decoded  = pooled + B * H;                // 16384 floats
    float* partials = decoded + B * H;               // 64 * 16384 floats

    pool_partial_kernel<<<dim3(B, H / 256, POOL_CHUNKS), 256, 0, stream>>>(hidden, partials);
    pool_finalize_kernel<<<(B * H) / 256, 256, 0, stream>>>(partials, pooled);
    brain_kernel<<<1, 512, 0, stream>>>(pooled, gumbel_u,
                                        enc_w1, enc_b1, ln1_g, ln1_b, enc_w2, enc_b2,
                                        embed, dec_w1, dec_b1, ln2_g, ln2_b, dec_w2, dec_b2,
                                        decoded);
    const unsigned total4 = (unsigned)B * S * (H / 4);   // 16M float4
    residual_kernel<<<8192, 256, 0, stream>>>((const float4*)hidden, (const float4*)decoded,
                                              (float4*)d_out, total4);
}